// AtomEncoderWithCoordDistCutoff_57887569215661
// MI455X (gfx1250) — compile-verified
//
#include <hip/hip_runtime.h>
#include <hip/hip_bf16.h>

// Shapes from the reference
#define NPROT 4
#define LRES  256
#define NATOM 14
#define MTOT  (LRES * NATOM)      // 3584
#define TILES (MTOT / 16)         // 224 tiles per dimension
#define JGRP  (TILES / 4)         // 56 groups of 4 j-tiles (16x64 strip per wave)
#define MAXD2 100.0f              // MAX_DIST^2

typedef __attribute__((ext_vector_type(2))) float v2f;
typedef __attribute__((ext_vector_type(8))) float v8f;

// ---------------------------------------------------------------------------
// Kernel 1: feats (N, M, 128) + mask (N, M)
// One wave per atom; lane writes channels [4*lane .. 4*lane+3] as a b128 store.
// Channels: [0..61] res_emb[aa], [62..124] atom_emb[atom_id], [125..127] coords.
// ---------------------------------------------------------------------------
__global__ __launch_bounds__(256) void feats_mask_kernel(
    const int* __restrict__ aa,          // (N, L)
    const float* __restrict__ pos14,     // (N, L, 14, 3) == (N, M, 3)
    const int* __restrict__ atom_mask,   // (N, M)
    const float* __restrict__ res_emb,   // (21, 62)
    const float* __restrict__ atom_emb,  // (14, 63)
    float* __restrict__ feats,           // (N, M, 128)
    float* __restrict__ mask)            // (N, M)
{
    const int wave = (blockIdx.x * blockDim.x + threadIdx.x) >> 5;  // global atom id
    const int lane = threadIdx.x & 31;
    const int n = wave / MTOT;
    const int m = wave - n * MTOT;
    const int l = m / NATOM;
    const int at = m - l * NATOM;

    const int rid = aa[n * LRES + l];
    const float* __restrict__ rrow = res_emb + rid * 62;
    const float* __restrict__ arow = atom_emb + at * 63;
    const float* __restrict__ crow = pos14 + (size_t)wave * 3;

    const int c0 = lane * 4;
    float v[4];
#pragma unroll
    for (int k = 0; k < 4; ++k) {
        const int c = c0 + k;
        float val;
        if (c < 62)       val = rrow[c];
        else if (c < 125) val = arow[c - 62];
        else              val = crow[c - 125];
        v[k] = val;
    }
    float4 o = make_float4(v[0], v[1], v[2], v[3]);
    *(float4*)(feats + (size_t)wave * 128 + c0) = o;   // global_store_b128

    if (lane == 0)
        mask[wave] = atom_mask[wave] ? 1.0f : 0.0f;
}

// ---------------------------------------------------------------------------
// Kernel 2: edge_mask (N, M, M) via WMMA with the row-norm FOLDED INTO K=4:
//   A_hat[i] = (-2x_i, -2y_i, -2z_i, |c_i|^2),  B_hat[j] = (x_j, y_j, z_j, 1)
//   => D[i][j] = |c_i|^2 - 2 c_i.c_j ;  dist^2 = D + |c_j|^2
// |c_j|^2 is indexed by COLUMN == the per-lane index of the D layout, so the
// epilogue is a plain per-lane add.
//
// Each lane loads its row's full 12-byte coordinate triple (mergeable into
// global_load_b96) and computes the norm LOCALLY: no cross-lane shuffles, no
// ds ops, and every load is unconditionally used so the compiler emits pure
// v_cndmask operand selects instead of divergent EXEC branches (EXEC stays
// all-ones, as WMMA requires).
//
// One wave computes a 16x64 strip: A_hat built once, 4 WMMAs over 4 j-tiles.
// All 32 stores share one address register (immediate offsets < 2^17 bytes).
//
// A (16x4 f32) per-lane layout (ISA 7.12.2): lane = h*16 + t holds
//   vgpr0 = A[t][2h], vgpr1 = A[t][2h+1];  B (4x16) mirrors it per column.
// D (16x16 f32): vgpr v, lane h*16+t -> element (v + 8h, t).
// ---------------------------------------------------------------------------
__global__ __launch_bounds__(256) void edge_mask_kernel(
    const float* __restrict__ pos14,   // (N, M, 3)
    float* __restrict__ edge)          // (N, M, M)
{
    const int lane = threadIdx.x & 31;
    const int w = blockIdx.x * 8 + (threadIdx.x >> 5);
    // total strips = NPROT * TILES * JGRP = 4*224*56 = 50176 ; grid = 6272 exact
    const int n    = w / (TILES * JGRP);
    const int rem  = w - n * (TILES * JGRP);
    const int ti   = rem / JGRP;
    const int tjg  = rem - ti * JGRP;
    const int i0 = ti * 16;
    const int jb = tjg * 64;

    const int t = lane & 15;
    const int h = lane >> 4;

    const float* __restrict__ C = pos14 + (size_t)n * MTOT * 3;

    // ---- Build A_hat once (rows i0..i0+15): full-row load, local norm ----
    const float* __restrict__ ra = C + (size_t)(i0 + t) * 3;
    const float xa = ra[0], ya = ra[1], za = ra[2];      // 3 dwords, mergeable
    const float na = xa * xa + ya * ya + za * za;        // |c_i|^2, per lane
    v2f a;
    a.x = h ? (-2.0f * za) : (-2.0f * xa);
    a.y = h ? na           : (-2.0f * ya);

    // ---- Output base: element (i0, jb) ----
    float* __restrict__ outp =
        edge + ((size_t)n * MTOT + i0) * MTOT + jb + t;

#pragma unroll
    for (int jt = 0; jt < 4; ++jt) {
        const float* __restrict__ rb = C + (size_t)(jb + jt * 16 + t) * 3;
        const float xb = rb[0], yb = rb[1], zb = rb[2];
        const float nb = xb * xb + yb * yb + zb * zb;    // |c_j|^2, column lane
        v2f b;
        b.x = h ? zb   : xb;
        b.y = h ? 1.0f : yb;

        v8f acc = {};
        acc = __builtin_amdgcn_wmma_f32_16x16x4_f32(
            false, a, false, b, (short)0, acc, false, false);

#pragma unroll
        for (int v = 0; v < 8; ++v) {
            const float d2  = acc[v] + nb;               // dist^2
            const float val = (d2 < MAXD2) ? 1.0f : 0.0f;
            // row (v + 8h), column offset jt*16: single addr reg + immediates
            outp[(size_t)(v + 8 * h) * MTOT + jt * 16] = val;
        }
    }
}

// ---------------------------------------------------------------------------
// d_out layout (flat concat, all f32): feats | mask | edge_mask
// ---------------------------------------------------------------------------
extern "C" void kernel_launch(void* const* d_in, const int* in_sizes, int n_in,
                              void* d_out, int out_size, void* d_ws, size_t ws_size,
                              hipStream_t stream) {
    const int*   aa        = (const int*)d_in[0];
    const float* pos14     = (const float*)d_in[1];
    const int*   atom_mask = (const int*)d_in[2];
    const float* res_emb   = (const float*)d_in[3];
    const float* atom_emb  = (const float*)d_in[4];

    float* out   = (float*)d_out;
    float* feats = out;                                       // N*M*128
    float* mask  = out + (size_t)NPROT * MTOT * 128;          // N*M
    float* edge  = mask + (size_t)NPROT * MTOT;               // N*M*M

    // Kernel 1: one wave per atom. 4*3584 atoms -> 14336 waves / 8 per block.
    feats_mask_kernel<<<(NPROT * MTOT) / 8, 256, 0, stream>>>(
        aa, pos14, atom_mask, res_emb, atom_emb, feats, mask);

    // Kernel 2: one wave per 16x64 strip. 4*224*56 strips / 8 per block.
    edge_mask_kernel<<<(NPROT * TILES * JGRP) / 8, 256, 0, stream>>>(pos14, edge);
}